// UnfoldindAndAttention_40381282517564
// MI455X (gfx1250) — compile-verified
//
#include <hip/hip_runtime.h>

typedef __attribute__((ext_vector_type(2))) float v2f;
typedef __attribute__((ext_vector_type(8))) float v8f;

#define D_FEAT 64
#define ALP 0.5f
#define TAU 0.2f
#define T_THRESH 2.0f

// ---------------- degree count: deg[dst[e]] += 1 ----------------
__global__ void deg_count_kernel(const int* __restrict__ dst,
                                 float* __restrict__ deg, int E) {
  int e = blockIdx.x * blockDim.x + threadIdx.x;
  if (e < E) atomicAdd(deg + dst[e], 1.0f);
}

// ---------------- per-node norms: rs = (deg+1)^-1/2, rinv = (deg+1)^-1 ----------------
__global__ void node_prep_kernel(const float* __restrict__ deg,
                                 float* __restrict__ rs,
                                 float* __restrict__ rinv, int N) {
  int i = blockIdx.x * blockDim.x + threadIdx.x;
  if (i < N) {
    float t = deg[i] + 1.0f;            // LAM = 1
    rs[i]   = rsqrtf(t);
    rinv[i] = 1.0f / t;
  }
}

// ---------------- edge scatter: agg[dst] += Y[src] * rs[src] * w ----------------
// one thread per (edge, 4-float chunk): 16 threads/edge, float4 loads, f32 atomics
__global__ void scatter_kernel(const float* __restrict__ Yin,
                               const float* __restrict__ rs,
                               const float* __restrict__ w,
                               const int* __restrict__ src,
                               const int* __restrict__ dst,
                               float* __restrict__ agg, int E, int useW) {
  long long tid = (long long)blockIdx.x * blockDim.x + threadIdx.x;
  int e = (int)(tid >> 4);
  int c = ((int)tid) & 15;
  if (e >= E) return;
  int s = src[e];
  int d = dst[e];
  float wg = rs[s] * (useW ? w[e] : 1.0f);
  const float4 y = *(const float4*)(Yin + (long long)s * D_FEAT + c * 4);
  float* a = agg + (long long)d * D_FEAT + c * 4;
  atomicAdd(a + 0, y.x * wg);
  atomicAdd(a + 1, y.y * wg);
  atomicAdd(a + 2, y.z * wg);
  atomicAdd(a + 3, y.w * wg);
}

// ---------------- node combine: Yout = 0.5*Yin + 0.5*rs*agg + 0.5*rinv*X ----------------
__global__ void combine_kernel(const float* __restrict__ Yin,
                               const float* __restrict__ agg,
                               const float* __restrict__ X,
                               const float* __restrict__ rs,
                               const float* __restrict__ rinv,
                               float* __restrict__ Yout, int N) {
  long long tid = (long long)blockIdx.x * blockDim.x + threadIdx.x;
  int i = (int)(tid >> 4);
  int c = ((int)tid) & 15;
  if (i >= N) return;
  float si = ALP * rs[i];               // ALP*LAM*rs
  float ri = ALP * rinv[i];
  long long off = (long long)i * D_FEAT + c * 4;
  float4 y = *(const float4*)(Yin + off);
  float4 g = *(const float4*)(agg + off);
  float4 x = *(const float4*)(X + off);
  float4 o;
  o.x = (1.0f - ALP) * y.x + si * g.x + ri * x.x;
  o.y = (1.0f - ALP) * y.y + si * g.y + ri * x.y;
  o.z = (1.0f - ALP) * y.z + si * g.z + ri * x.z;
  o.w = (1.0f - ALP) * y.w + si * g.w + ri * x.w;
  *(float4*)(Yout + off) = o;
}

// ---------------- attention: nsq per edge via WMMA Gram diag ----------------
// One wave handles 16 edges. M[row,d] = etas[d]*(Y[src,d]-Y[dst,d]) (16x64).
// diag(M M^T) accumulated with v_wmma_f32_16x16x4_f32 over 16 K=4 chunks.
// A(16x4) layout == B(4x16) layout for M and M^T, so A and B share registers.
__global__ void attention_wmma_kernel(const float* __restrict__ Y,
                                      const float* __restrict__ etas,
                                      const int* __restrict__ src,
                                      const int* __restrict__ dst,
                                      float* __restrict__ w,
                                      float* __restrict__ deg, int E) {
  int lane = threadIdx.x & 31;
  long long waveId = (long long)blockIdx.x * (blockDim.x >> 5) + (threadIdx.x >> 5);
  long long e0 = waveId * 16;
  if (e0 >= E) return;                  // wave-uniform: EXEC stays all-ones

  int row  = lane & 15;                 // edge row within the 16
  int koff = (lane >> 4) * 2;           // lanes 16-31 hold K=2,3 of each chunk
  long long eRow = e0 + row;
  int eC = (int)(eRow < E ? eRow : (long long)E - 1);   // clamp keeps EXEC full
  const float* ys = Y + (long long)src[eC] * D_FEAT;
  const float* yd = Y + (long long)dst[eC] * D_FEAT;

  v8f acc = {};
#pragma unroll
  for (int c = 0; c < 16; ++c) {
    int d0 = 4 * c + koff;
    v2f a;
    a.x = etas[d0]     * (ys[d0]     - yd[d0]);
    a.y = etas[d0 + 1] * (ys[d0 + 1] - yd[d0 + 1]);
    // D = A*A^T + C  (same regs feed A and B)
    acc = __builtin_amdgcn_wmma_f32_16x16x4_f32(
        false, a, false, a, (short)0, acc, false, false);
  }

  // diagonal (m,m): m<8 -> lane m, VGPR m ; m>=8 -> lane m+16, VGPR m-8
  bool hasDiag = (lane < 8) || (lane >= 24);
  int m = (lane < 8) ? lane : (lane - 16);
  int r = (lane < 8) ? lane : (lane - 24);
  float nsq = 0.0f;
#pragma unroll
  for (int q = 0; q < 8; ++q)
    if (r == q) nsq = acc[q];

  long long eIdx = e0 + m;
  if (hasDiag && eIdx < E) {
    float wv = 0.5f * rsqrtf(nsq + 1e-7f);   // 0.5*P*(nsq+eps)^(0.5P-1), P=1
    if (nsq > T_THRESH) wv = TAU;
    w[eIdx] = wv;
    atomicAdd(deg + dst[(int)eIdx], wv);
  }
}

static inline size_t alignUp(size_t x, size_t a) { return (x + a - 1) & ~(a - 1); }

extern "C" void kernel_launch(void* const* d_in, const int* in_sizes, int n_in,
                              void* d_out, int out_size, void* d_ws, size_t ws_size,
                              hipStream_t stream) {
  const float* X    = (const float*)d_in[0];
  const float* etas = (const float*)d_in[1];
  const int*   src  = (const int*)d_in[2];
  const int*   dst  = (const int*)d_in[3];
  const int N = in_sizes[0] / D_FEAT;
  const int E = in_sizes[2];
  float* Yout = (float*)d_out;

  // workspace carve-up (~32.6 MB): two ping-pong node buffers + w + small arrays
  size_t nodeBytes = (size_t)N * D_FEAT * sizeof(float);
  char* p = (char*)d_ws;
  float* B1   = (float*)p; p += alignUp(nodeBytes, 256);
  float* B2   = (float*)p; p += alignUp(nodeBytes, 256);
  float* wbuf = (float*)p; p += alignUp((size_t)E * sizeof(float), 256);
  float* deg  = (float*)p; p += alignUp((size_t)N * sizeof(float), 256);
  float* rs   = (float*)p; p += alignUp((size_t)N * sizeof(float), 256);
  float* rinv = (float*)p; p += alignUp((size_t)N * sizeof(float), 256);

  const int TB = 256;

  // initial (unweighted) in-degrees and norms
  hipMemsetAsync(deg, 0, (size_t)N * sizeof(float), stream);
  deg_count_kernel<<<(E + TB - 1) / TB, TB, 0, stream>>>(dst, deg, E);
  node_prep_kernel<<<(N + TB - 1) / TB, TB, 0, stream>>>(deg, rs, rinv, N);

  const float* Yin = X;                // Y0 = X, no copy
  float* bufs[2] = {B1, B2};
  long long edgeThreads = (long long)E * 16;
  long long nodeThreads = (long long)N * 16;
  int edgeGrid = (int)((edgeThreads + TB - 1) / TB);
  int nodeGrid = (int)((nodeThreads + TB - 1) / TB);

  for (int k = 0; k < 8; ++k) {
    float* agg = bufs[k & 1];          // never equals Yin (alternation)
    hipMemsetAsync(agg, 0, nodeBytes, stream);
    scatter_kernel<<<edgeGrid, TB, 0, stream>>>(Yin, rs, wbuf, src, dst, agg,
                                                E, (k >= 4) ? 1 : 0);
    float* outBuf = (k == 7) ? Yout : agg;   // combine in-place into agg
    combine_kernel<<<nodeGrid, TB, 0, stream>>>(Yin, agg, X, rs, rinv, outBuf, N);
    Yin = outBuf;

    if (k == 3) {                      // attention after 4th propagation
      hipMemsetAsync(deg, 0, (size_t)N * sizeof(float), stream);
      attention_wmma_kernel<<<(E + 127) / 128, TB, 0, stream>>>(
          Yin, etas, src, dst, wbuf, deg, E);
      node_prep_kernel<<<(N + TB - 1) / TB, TB, 0, stream>>>(deg, rs, rinv, N);
    }
  }
}